// GraphConvolution_38311108280995
// MI455X (gfx1250) — compile-verified
//
#include <hip/hip_runtime.h>
#include <cstddef>

// ---------------------------------------------------------------------------
// GraphConvolution for MI455X (gfx1250, wave32, WMMA)
//
// Stage 1:  support = X @ W^T + b_lin          [N,256] -- fp32 WMMA 16x16x4
// Stage 2:  out = tanh(sum_k support[a2a] * fedges[n2e]) + bias   -- BW bound
// ---------------------------------------------------------------------------

typedef __attribute__((ext_vector_type(2))) float v2f;
typedef __attribute__((ext_vector_type(8))) float v8f;

#define DIM     256
#define MAX_NEI 16

// ---------------------------------------------------------------------------
// Kernel 1: fp32 WMMA GEMM.  One wave -> one 16x16 tile of `support`.
//
// VGPR layouts (CDNA5 ISA 7.12.2, 32-bit operands, wave32):
//   A 16x4 (MxK):  lanes 0-15 hold row M=lane, K={k,k+1}; lanes 16-31 hold
//                  row M=lane-16, K={k+2,k+3}  -> contiguous float2 load.
//   B 4x16 (KxN):  mirrored: lane holds col N=lane&15, same K split
//                  -> B[kk][n] = W[n][kk], contiguous float2 from W row.
//   C/D 16x16:     VGPR i: lanes 0-15 -> M=i, lanes 16-31 -> M=i+8.
// ---------------------------------------------------------------------------
__global__ __launch_bounds__(256) void support_gemm_wmma(
    const float* __restrict__ X,      // [N, 256]
    const float* __restrict__ W,      // [256, 256] (row o, col d)
    const float* __restrict__ b_lin,  // [256]
    float* __restrict__ support,      // [N, 256]  (workspace)
    int nNodes)
{
    const int lane  = threadIdx.x & 31;
    const int wave  = threadIdx.x >> 5;        // 0..7
    const int laneM = lane & 15;               // row (A) / col (B) within tile
    const int kHi   = lane >> 4;               // 0: K={0,1}, 1: K={2,3}

    const int row0 = blockIdx.x * 16;                  // node band
    const int col0 = (blockIdx.y * 8 + wave) * 16;     // output-channel tile
    if (row0 >= nNodes) return;

    const float* __restrict__ arow = X + (size_t)(row0 + laneM) * DIM + 2 * kHi;
    const float* __restrict__ brow = W + (size_t)(col0 + laneM) * DIM + 2 * kHi;

    v8f acc = {};
#pragma unroll 8
    for (int k = 0; k < DIM; k += 4) {
        v2f a = *(const v2f*)(arow + k);   // A[M=laneM][k+2*kHi .. +1]
        v2f b = *(const v2f*)(brow + k);   // B[k+2*kHi .. +1][N=laneM] = W[col][d]
        // 8 args: (neg_a, A, neg_b, B, c_mod, C, reuse_a, reuse_b)
        acc = __builtin_amdgcn_wmma_f32_16x16x4_f32(
            false, a, false, b, (short)0, acc, false, false);
    }

    const float bo = b_lin[col0 + laneM];
    float* __restrict__ orow =
        support + (size_t)(row0 + 8 * kHi) * DIM + col0 + laneM;
#pragma unroll
    for (int i = 0; i < 8; ++i) {
        orow[(size_t)i * DIM] = acc[i] + bo;   // row = row0 + i + 8*kHi
    }
}

// ---------------------------------------------------------------------------
// Kernel 2: gather + elementwise multiply + K-reduce + tanh + bias.
// 64 lanes per node, float4 (b128) coalesced row loads. Pure bandwidth.
// ---------------------------------------------------------------------------
__global__ __launch_bounds__(256) void gather_mul_reduce(
    const float* __restrict__ support,   // [N, 256]
    const float* __restrict__ fedges,    // [E, 256]
    const long long* __restrict__ a2a,   // [N, 16] int64
    const long long* __restrict__ n2e,   // [N, 16] int64
    const float* __restrict__ bias,      // [256]
    float* __restrict__ out,             // [N, 256]
    int nNodes)
{
    const int node = blockIdx.x * 4 + (threadIdx.x >> 6);  // 4 nodes / block
    const int d    = (threadIdx.x & 63) * 4;               // 4 channels / lane
    if (node >= nNodes) return;

    float4 acc = make_float4(0.f, 0.f, 0.f, 0.f);
#pragma unroll
    for (int k = 0; k < MAX_NEI; ++k) {
        const long long ni = a2a[(size_t)node * MAX_NEI + k];
        const long long ei = n2e[(size_t)node * MAX_NEI + k];
        const float4 s = *(const float4*)(support + (size_t)ni * DIM + d);
        const float4 f = *(const float4*)(fedges  + (size_t)ei * DIM + d);
        acc.x += s.x * f.x;
        acc.y += s.y * f.y;
        acc.z += s.z * f.z;
        acc.w += s.w * f.w;
    }

    const float4 b = *(const float4*)(bias + d);
    float4 r;
    r.x = tanhf(acc.x) + b.x;
    r.y = tanhf(acc.y) + b.y;
    r.z = tanhf(acc.z) + b.z;
    r.w = tanhf(acc.w) + b.w;
    *(float4*)(out + (size_t)node * DIM + d) = r;
}

// ---------------------------------------------------------------------------
// Launch. Inputs (setup_inputs order):
//   0 input_features [N,256] f32   1 fedges [E,256] f32
//   2 a2a [N,16] i64               3 node2edge [N,16] i64
//   4 W [256,256] f32              5 b_lin [256] f32       6 bias [256] f32
// d_ws holds `support` (N*256 floats = 51.2 MB).
// ---------------------------------------------------------------------------
extern "C" void kernel_launch(void* const* d_in, const int* in_sizes, int n_in,
                              void* d_out, int out_size, void* d_ws, size_t ws_size,
                              hipStream_t stream)
{
    const float*     X      = (const float*)d_in[0];
    const float*     fedges = (const float*)d_in[1];
    const long long* a2a    = (const long long*)d_in[2];
    const long long* n2e    = (const long long*)d_in[3];
    const float*     W      = (const float*)d_in[4];
    const float*     b_lin  = (const float*)d_in[5];
    const float*     bias   = (const float*)d_in[6];
    float*           out    = (float*)d_out;
    float*           support = (float*)d_ws;

    const int nNodes = in_sizes[0] / DIM;   // 50000 (multiple of 16)

    // Stage 1: WMMA GEMM. grid = (N/16 node bands) x (2 blocks of 8 col-tiles)
    dim3 g1((nNodes + 15) / 16, DIM / (8 * 16));
    support_gemm_wmma<<<g1, 256, 0, stream>>>(X, W, b_lin, support, nNodes);

    // Stage 2: gather / reduce / tanh. 4 nodes per 256-thread block.
    dim3 g2((nNodes + 3) / 4);
    gather_mul_reduce<<<g2, 256, 0, stream>>>(support, fedges, a2a, n2e,
                                              bias, out, nNodes);
}